// MultiheadAdditiveAttention_13365938225758
// MI455X (gfx1250) — compile-verified
//
#include <hip/hip_runtime.h>

#define B_ 4
#define T_ 128
#define S_ 512
#define A_ 512

typedef __attribute__((ext_vector_type(2))) float v2f;
typedef __attribute__((ext_vector_type(8))) float v8f;

// tanh(x) = (e^{2x}-1)/(e^{2x}+1), 2 trans ops (v_exp_f32 + v_rcp_f32).
// Clamp to +-9: tanh(9) == 1.0f in fp32, avoids inf/inf NaN.
__device__ __forceinline__ float fast_tanh(float x) {
    x = fminf(9.0f, fmaxf(-9.0f, x));
    float t = __builtin_amdgcn_exp2f(x * 2.88539008177792681472f); // 2*log2(e)
    return (t - 1.0f) * __builtin_amdgcn_rcpf(t + 1.0f);
}

__device__ __forceinline__ float fast_exp(float x) {
    return __builtin_amdgcn_exp2f(x * 1.44269504088896340736f);
}

// ---------------------------------------------------------------------------
// Y[M,N] = X[M,K] @ W[N,K]^T + bias[N], K == N == 512. One wave per 16x16
// output tile, f32 WMMA 16x16x4 along K.
//
// A operand (16x4 f32, ISA 7.12.2): lane L holds row M = L&15;
//   lanes 0-15: VGPR0=K0, VGPR1=K1; lanes 16-31: VGPR0=K2, VGPR1=K3.
// B operand (4x16 f32): lane L holds col N = L&15, same K-pair split.
// C/D: VGPR r -> (M=r, N=lane) lanes 0-15; (M=r+8, N=lane-16) lanes 16-31.
// ---------------------------------------------------------------------------
__global__ void proj_gemm(const float* __restrict__ X, const float* __restrict__ W,
                          const float* __restrict__ bias, float* __restrict__ Y,
                          int M) {
    const int K = A_, N = A_;
    const int wave = threadIdx.x >> 5;
    const int lane = threadIdx.x & 31;
    const int tile = blockIdx.x * 4 + wave;
    const int ntn  = N / 16;
    const int mt = tile / ntn, nt = tile % ntn;
    if (mt * 16 >= M) return;                  // wave-uniform; EXEC stays all-1s
    const int m0 = mt * 16, n0 = nt * 16;
    const int r15   = lane & 15;
    const int khalf = (lane >> 4) * 2;

    const float* xrow = X + (long)(m0 + r15) * K + khalf;
    const float* wrow = W + (long)(n0 + r15) * K + khalf;

    v8f acc = {};
#pragma unroll 4
    for (int k0 = 0; k0 < K; k0 += 4) {
        v2f a = *(const v2f*)(xrow + k0);
        v2f b = *(const v2f*)(wrow + k0);
        acc = __builtin_amdgcn_wmma_f32_16x16x4_f32(false, a, false, b,
                                                    (short)0, acc, false, false);
    }

    const float bv    = bias[n0 + r15];
    const int   rbase = m0 + ((lane >> 4) << 3);
    const int   col   = n0 + r15;
#pragma unroll
    for (int r = 0; r < 8; ++r)
        Y[(long)(rbase + r) * N + col] = acc[r] + bv;
}

// ---------------------------------------------------------------------------
// Fused additive-attention scores + softmax. One block (256 thr = 8 waves)
// per (b,t) row. score[s] = sum_a va[a]*tanh(q[a]+k[b,s,a]) + va_b + mask.
// This is the dominant stage: B*T*S*A = 134M tanh evaluations (VALU-bound).
// ---------------------------------------------------------------------------
__global__ void score_softmax(const float* __restrict__ Q,   // [B*T, A]
                              const float* __restrict__ Kp,  // [B*S, A]
                              const float* __restrict__ mask,// [B*T, S]
                              const float* __restrict__ va,  // [A]
                              const float* __restrict__ vab, // [1]
                              float* __restrict__ Wout) {    // [B*T, S]
    __shared__ float qrow[A_];
    __shared__ float vaw[A_];
    __shared__ float sc[S_];
    __shared__ float red[256];

    const int bt   = blockIdx.x;      // 0 .. B*T-1
    const int b    = bt / T_;
    const int tid  = threadIdx.x;
    const int lane = tid & 31;
    const int wave = tid >> 5;

    for (int i = tid; i < A_; i += 256) {
        qrow[i] = Q[(long)bt * A_ + i];
        vaw[i]  = va[i];
    }
    __syncthreads();

    const float vb = vab[0];
    for (int s = wave; s < S_; s += 8) {
        const float* krow = Kp + (long)(b * S_ + s) * A_;
        float acc = 0.f;
#pragma unroll
        for (int i = 0; i < A_ / 128; ++i) {   // lanes sweep 128 elems per iter
            int a = i * 128 + lane * 4;
            float4 kk = *(const float4*)(krow + a);
            float4 qq = *(const float4*)(qrow + a);
            float4 vv = *(const float4*)(vaw + a);
            acc += vv.x * fast_tanh(qq.x + kk.x);
            acc += vv.y * fast_tanh(qq.y + kk.y);
            acc += vv.z * fast_tanh(qq.z + kk.z);
            acc += vv.w * fast_tanh(qq.w + kk.w);
        }
#pragma unroll
        for (int off = 16; off > 0; off >>= 1)
            acc += __shfl_xor(acc, off, 32);
        if (lane == 0)
            sc[s] = acc + vb + mask[(long)bt * S_ + s];
    }
    __syncthreads();

    // ---- softmax over sc[0..511], 2 elements per thread ----
    float m = fmaxf(sc[tid], sc[tid + 256]);
    red[tid] = m;
    __syncthreads();
    for (int off = 128; off > 0; off >>= 1) {
        if (tid < off) red[tid] = fmaxf(red[tid], red[tid + off]);
        __syncthreads();
    }
    m = red[0];
    __syncthreads();

    float e0 = fast_exp(sc[tid] - m);
    float e1 = fast_exp(sc[tid + 256] - m);
    red[tid] = e0 + e1;
    __syncthreads();
    for (int off = 128; off > 0; off >>= 1) {
        if (tid < off) red[tid] += red[tid + off];
        __syncthreads();
    }
    const float inv = 1.0f / red[0];
    Wout[(long)bt * S_ + tid]       = e0 * inv;
    Wout[(long)bt * S_ + tid + 256] = e1 * inv;
}

// ---------------------------------------------------------------------------
// out[b,t,n] = sum_s w[b,t,s] * values[b,s,n]. One wave per 16x16 tile,
// f32 WMMA 16x16x4 along S.
// ---------------------------------------------------------------------------
__global__ void attn_out_gemm(const float* __restrict__ Wt,  // [B,T,S]
                              const float* __restrict__ V,   // [B,S,A]
                              float* __restrict__ out) {     // [B,T,A]
    const int wave = threadIdx.x >> 5;
    const int lane = threadIdx.x & 31;
    const int tile = blockIdx.x * 4 + wave;
    const int NT = T_ / 16, NN = A_ / 16;
    const int b   = tile / (NT * NN);
    const int rem = tile % (NT * NN);
    const int mt = rem / NN, nt = rem % NN;
    const int m0 = mt * 16, n0 = nt * 16;
    const int r15   = lane & 15;
    const int khalf = (lane >> 4) * 2;

    const float* arow  = Wt + (long)(b * T_ + m0 + r15) * S_ + khalf;
    const float* vbase = V + (long)b * S_ * A_ + (n0 + r15);

    v8f acc = {};
#pragma unroll 2
    for (int s0 = 0; s0 < S_; s0 += 4) {
        v2f a = *(const v2f*)(arow + s0);
        v2f bv;
        bv[0] = vbase[(long)(s0 + khalf)     * A_];
        bv[1] = vbase[(long)(s0 + khalf + 1) * A_];
        acc = __builtin_amdgcn_wmma_f32_16x16x4_f32(false, a, false, bv,
                                                    (short)0, acc, false, false);
    }

    const int rbase = m0 + ((lane >> 4) << 3);
    const int col   = n0 + r15;
#pragma unroll
    for (int r = 0; r < 8; ++r)
        out[(long)(b * T_ + rbase + r) * A_ + col] = acc[r];
}

extern "C" void kernel_launch(void* const* d_in, const int* in_sizes, int n_in,
                              void* d_out, int out_size, void* d_ws, size_t ws_size,
                              hipStream_t stream) {
    const float* queries = (const float*)d_in[0];
    const float* keys    = (const float*)d_in[1];
    const float* values  = (const float*)d_in[2];
    const float* mask    = (const float*)d_in[3];
    const float* Wa_w    = (const float*)d_in[4];
    const float* Wa_b    = (const float*)d_in[5];
    const float* Ua_w    = (const float*)d_in[6];
    const float* Ua_b    = (const float*)d_in[7];
    const float* va_w    = (const float*)d_in[8];
    const float* va_b    = (const float*)d_in[9];
    float* out = (float*)d_out;

    float* kproj = (float*)d_ws;                       // B*S*A floats (4 MB)
    float* qproj = kproj + (size_t)B_ * S_ * A_;       // B*T*A floats (1 MB)
    float* wts   = qproj + (size_t)B_ * T_ * A_;       // B*T*S floats (1 MB)

    // k = keys @ Wa^T + Wa_b   (M = B*S = 2048): 4096 tiles, 4 waves/block
    proj_gemm<<<(B_ * S_ / 16) * (A_ / 16) / 4, 128, 0, stream>>>(
        keys, Wa_w, Wa_b, kproj, B_ * S_);
    // q = queries @ Ua^T + Ua_b (M = B*T = 512): 1024 tiles
    proj_gemm<<<(B_ * T_ / 16) * (A_ / 16) / 4, 128, 0, stream>>>(
        queries, Ua_w, Ua_b, qproj, B_ * T_);

    // dominant stage: fused tanh-score + softmax, one block per (b,t)
    score_softmax<<<B_ * T_, 256, 0, stream>>>(qproj, kproj, mask, va_w, va_b, wts);

    // out = w @ values: 1024 tiles
    attn_out_gemm<<<B_ * (T_ / 16) * (A_ / 16) / 4, 128, 0, stream>>>(
        wts, values, out);
}